// WaveletMask_74019466379564
// MI455X (gfx1250) — compile-verified
//
#include <hip/hip_runtime.h>
#include <hip/hip_bf16.h>

// CDNA5 / gfx1250, wave32. Fused Haar-DWT -> mask -> inverse as
//   y = A^T * ( W_C ⊙ (A * x * A^T) ) * A
// with A the orthonormal 128x128 Haar analysis operator, executed as four
// 128x128x128 f32 GEMMs on v_wmma_f32_16x16x4_f32. Single workgroup, all
// intermediates in LDS (2 x 64KB ping-pong), A generated analytically.

typedef __attribute__((ext_vector_type(2))) float v2f;
typedef __attribute__((ext_vector_type(8))) float v8f;

#define H 128
#define HB 64
#define NELEM (H * H)

// A[i,c]: row i of the Haar analysis matrix.
//  i <  64: low-pass  row: +r at c=2i, +r at c=2i+1
//  i >= 64: high-pass row: +r at c=2(i-64), -r at c=2(i-64)+1
__device__ __forceinline__ float haarA(int i, int c) {
  const float R = 0.70710678118654752440f; // 1/sqrt(2)
  bool lo = (i < HB);
  int ii = lo ? i : (i - HB);
  if ((c >> 1) != ii) return 0.0f;
  return (!lo && (c & 1)) ? -R : R;
}

// MODE: where the WMMA operands come from.
//  1: A-op = analytic A,   B-op = LDS S        (T  = A  * X )
//  2: A-op = LDS S,        B-op = analytic A^T (C  = T  * A^T)
//  3: A-op = analytic A^T, B-op = LDS S        (Z  = A^T* C')
//  4: A-op = LDS S,        B-op = analytic A   (Y  = Z  * A )
template <int MODE>
__device__ __forceinline__ void mm_stage(const float* __restrict__ S,
                                         float* __restrict__ D,
                                         int lane, int wave) {
  const int lm = lane & 15;
  const int kb = (lane < 16) ? 0 : 2; // K sub-column pair per lane half

#pragma unroll 2
  for (int t = 0; t < 8; ++t) {
    const int tile = wave * 8 + t; // 64 tiles over 8 waves
    const int I = tile >> 3;
    const int J = tile & 7;
    const int mg = I * 16 + lm; // A-operand row (this lane)
    const int ng = J * 16 + lm; // B-operand / C column (this lane)

    v8f acc = {};
#pragma unroll 4
    for (int k = 0; k < 32; ++k) {
      const int k0 = k * 4 + kb;
      v2f a, b;
      // ---- A operand: 16x4 tile, lane holds row mg, cols k0,k0+1 ----
      if (MODE == 1) {
        a[0] = haarA(mg, k0);
        a[1] = haarA(mg, k0 + 1);
      } else if (MODE == 3) { // A^T[m,c] = A[c,m]
        a[0] = haarA(k0, mg);
        a[1] = haarA(k0 + 1, mg);
      } else {
        a[0] = S[mg * H + k0];
        a[1] = S[mg * H + k0 + 1];
      }
      // ---- B operand: 4x16 tile, lane holds col ng, rows k0,k0+1 ----
      if (MODE == 2) { // A^T[k,n] = A[n,k]
        b[0] = haarA(ng, k0);
        b[1] = haarA(ng, k0 + 1);
      } else if (MODE == 4) {
        b[0] = haarA(k0, ng);
        b[1] = haarA(k0 + 1, ng);
      } else {
        b[0] = S[k0 * H + ng];
        b[1] = S[(k0 + 1) * H + ng];
      }
      acc = __builtin_amdgcn_wmma_f32_16x16x4_f32(
          /*neg_a=*/false, a, /*neg_b=*/false, b,
          /*c_mod=*/(short)0, acc, /*reuse_a=*/false, /*reuse_b=*/false);
    }

    // C/D layout: lane col = ng; rows = I*16 + vgpr + (lane<16 ? 0 : 8)
    const int rb = I * 16 + ((lane < 16) ? 0 : 8);
#pragma unroll
    for (int v = 0; v < 8; ++v) D[(rb + v) * H + ng] = acc[v];
  }
}

__global__ __launch_bounds__(256) void WaveletMask_kernel(
    const float* __restrict__ x, const float* __restrict__ w,
    float* __restrict__ y) {
  extern __shared__ float sm[];
  float* B0 = sm;         // 128*128 f32
  float* B1 = sm + NELEM; // 128*128 f32

  const int tid = threadIdx.x;
  const int lane = tid & 31;
  const int wave = tid >> 5;

  // Stage 0: x -> LDS
  for (int i = tid; i < NELEM; i += 256) B0[i] = x[i];
  __syncthreads();

  mm_stage<1>(B0, B1, lane, wave); // T = A * X
  __syncthreads();
  mm_stage<2>(B1, B0, lane, wave); // C = T * A^T  (= [[LL,HL],[LH,HH]])
  __syncthreads();

  // Mask: C' = C ⊙ W_C, where W_C swaps the off-diagonal weight quadrants
  // (C's top-right is HL, masked by weight[64+i, j]; bottom-left is LH,
  //  masked by weight[i, 64+j]).
  for (int i = tid; i < NELEM; i += 256) {
    const int r = i >> 7;
    const int c = i & (H - 1);
    int wr = r, wc = c;
    if (r < HB && c >= HB) { wr = r + HB; wc = c - HB; }
    else if (r >= HB && c < HB) { wr = r - HB; wc = c + HB; }
    B0[i] *= w[(wr << 7) + wc];
  }
  __syncthreads();

  mm_stage<3>(B0, B1, lane, wave); // Z = A^T * C'
  __syncthreads();
  mm_stage<4>(B1, y, lane, wave);  // Y = Z * A  -> global
}

extern "C" void kernel_launch(void* const* d_in, const int* in_sizes, int n_in,
                              void* d_out, int out_size, void* d_ws,
                              size_t ws_size, hipStream_t stream) {
  const float* x = (const float*)d_in[0];
  const float* w = (const float*)d_in[1];
  float* y = (float*)d_out;
  (void)in_sizes; (void)n_in; (void)out_size; (void)d_ws; (void)ws_size;

  const size_t smem = 2 * NELEM * sizeof(float); // 128 KB LDS (<< 320 KB/WGP)
  WaveletMask_kernel<<<dim3(1), dim3(256), smem, stream>>>(x, w, y);
}